// FuzzyficationLayer_8658654069382
// MI455X (gfx1250) — compile-verified
//
#include <hip/hip_runtime.h>

// Problem shape (from the reference)
#define NUM_INPUTS 4096
#define NUM_MF 64
#define OUT_ELEMS (NUM_INPUTS * NUM_MF)      // 262144 outputs
#define VEC 4                                 // outputs per thread
#define BLOCK 256                             // 8 wave32s per block
#define OUT_PER_BLOCK (BLOCK * VEC)           // 1024 outputs per block
#define X_PER_BLOCK (OUT_PER_BLOCK / NUM_MF)  // 16 distinct x values per block

// -0.5 * log2(e): exp(-0.5*t) == exp2(C*t)
#define NEG_HALF_LOG2E (-0.72134752044448170368f)

// Use the gfx1250 async global->LDS path when the toolchain declares it.
#if __has_builtin(__builtin_amdgcn_global_load_async_to_lds_b32) && \
    __has_builtin(__builtin_amdgcn_s_wait_asynccnt)
#define USE_ASYNC_LDS 1
#else
#define USE_ASYNC_LDS 0
#endif

typedef __attribute__((ext_vector_type(4))) float v4f;
typedef __attribute__((address_space(1))) int g_int;
typedef __attribute__((address_space(3))) int l_int;

__global__ __launch_bounds__(BLOCK) void FuzzyficationLayer_kernel(
    const float* __restrict__ x,          // [BATCH, NUM_INPUTS]; only row 0 used
    const float* __restrict__ params,     // interleaved (mi, sigma) pairs
    float* __restrict__ out)              // [OUT_ELEMS]
{
    __shared__ float lds_x[X_PER_BLOCK];

    const int bid = blockIdx.x;
    const int tid = threadIdx.x;
    const int iBase = bid * X_PER_BLOCK;   // first input index this block covers

    // Stage the 16 x-values this block reuses (each feeds 64 outputs) into LDS.
#if USE_ASYNC_LDS
    if (tid < X_PER_BLOCK) {
        __builtin_amdgcn_global_load_async_to_lds_b32(
            (g_int*)(x + iBase + tid),
            (l_int*)(&lds_x[tid]),
            /*offset=*/0, /*cpol=*/0);
    }
    __builtin_amdgcn_s_wait_asynccnt(0);   // no-op on waves that issued nothing
    __syncthreads();
#else
    if (tid < X_PER_BLOCK) lds_x[tid] = x[iBase + tid];
    __syncthreads();
#endif

    // Each thread produces 4 consecutive outputs; 64 % 4 == 0 so all 4 share
    // the same input index i.
    const int t  = bid * BLOCK + tid;
    const int k0 = t * VEC;                       // first flat output index
    const float xv = lds_x[(tid * VEC) >> 6];     // i within block

    // Two 16B loads cover 4 (mi, sigma) pairs: global_load_b128 x2.
    const v4f p0 = *(const v4f*)(params + 2 * k0);       // (m0,s0,m1,s1)
    const v4f p1 = *(const v4f*)(params + 2 * k0 + 4);   // (m2,s2,m3,s3)

    v4f r;
    {
        float d = xv - p0.x;
        r.x = __builtin_amdgcn_exp2f(d * d * p0.y * p0.y * NEG_HALF_LOG2E);
    }
    {
        float d = xv - p0.z;
        r.y = __builtin_amdgcn_exp2f(d * d * p0.w * p0.w * NEG_HALF_LOG2E);
    }
    {
        float d = xv - p1.x;
        r.z = __builtin_amdgcn_exp2f(d * d * p1.y * p1.y * NEG_HALF_LOG2E);
    }
    {
        float d = xv - p1.z;
        r.w = __builtin_amdgcn_exp2f(d * d * p1.w * p1.w * NEG_HALF_LOG2E);
    }

    // One 16B store: global_store_b128.
    *(v4f*)(out + k0) = r;
}

extern "C" void kernel_launch(void* const* d_in, const int* in_sizes, int n_in,
                              void* d_out, int out_size, void* d_ws, size_t ws_size,
                              hipStream_t stream) {
    (void)in_sizes; (void)n_in; (void)out_size; (void)d_ws; (void)ws_size;
    const float* x      = (const float*)d_in[0];   // [32, 4096] fp32
    const float* params = (const float*)d_in[1];   // [4096*64*64, 2] fp32 (first 4096*64 rows used)
    float* out          = (float*)d_out;           // [262144] fp32

    dim3 grid(OUT_ELEMS / OUT_PER_BLOCK);          // 256 blocks
    dim3 block(BLOCK);                             // 256 threads = 8 waves
    FuzzyficationLayer_kernel<<<grid, block, 0, stream>>>(x, params, out);
}